// GAT_40037685133531
// MI455X (gfx1250) — compile-verified
//
#include <hip/hip_runtime.h>
#include <hip/hip_bf16.h>
#include <math.h>

// ---------------- problem constants (from reference) ----------------
#define N_NODES 50000
#define N_EDGES 800000
#define F_IN    128
#define HIDDEN  32
#define HEADS   4
#define HC      (HEADS * HIDDEN)   // 128
#define F_OUT   64
#define NEG_SLOPE 0.2f

typedef float v2f __attribute__((ext_vector_type(2)));
typedef float v8f __attribute__((ext_vector_type(8)));

// ---------------- f32 WMMA GEMM: C[M,N] = A[M,K] @ B[K,N] (+ bias[n]) ----
// Compile-time N,K -> all strides fold into immediate offsets.
// One wave32 per 16x16 output tile, 8 waves/block sharing one B column
// tile staged in LDS (tilesM padded to 8 so tn is block-uniform).
template <int N, int K, bool USE_BIAS>
__global__ __launch_bounds__(256) void gat_gemm_f32_wmma(
    const float* __restrict__ A, const float* __restrict__ B,
    const float* __restrict__ bias, float* __restrict__ C, int M) {
    __shared__ float ldsB[K * 16];     // 16-wide B column tile (8 KB for K=128)

    const int lane = threadIdx.x & 31;
    const int wib  = threadIdx.x >> 5;           // wave in block: 0..7
    const int tilesM    = M >> 4;
    const int tilesMpad = (tilesM + 7) & ~7;     // multiple of 8
    const int tile = blockIdx.x * 8 + wib;
    const int tn   = tile / tilesMpad;           // block-uniform by construction
    const int tm   = tile - tn * tilesMpad;

    // cooperative load of B[:, tn*16 .. +16] into LDS
    const float* Bcol = B + tn * 16;
#pragma unroll
    for (int i = threadIdx.x; i < K * 16; i += 256)
        ldsB[i] = Bcol[(i >> 4) * N + (i & 15)];
    __syncthreads();

    if (tm >= tilesM) return;                    // wave-uniform, after barrier

    const int hi    = lane >> 4;                 // lane half: 0/1
    const int lo    = lane & 15;
    const int khalf = hi * 2;                    // K sub-offset for this half

    const float* Arow = A + (size_t)(tm * 16 + lo) * K + khalf;
    const float* Blds = ldsB + khalf * 16 + lo;

    v8f acc = {};
#pragma unroll
    for (int k0 = 0; k0 < K; k0 += 4) {
        v2f a = *(const v2f*)(Arow + k0);        // b64 load, immediate offset
        v2f b;
        b.x = Blds[(k0)     * 16];               // ds loads, immediate offsets
        b.y = Blds[(k0 + 1) * 16];
        acc = __builtin_amdgcn_wmma_f32_16x16x4_f32(
            /*neg_a=*/false, a, /*neg_b=*/false, b,
            /*c_mod=*/(short)0, acc, /*reuse_a=*/false, /*reuse_b=*/false);
    }

    float badd = 0.0f;
    if (USE_BIAS) badd = bias[tn * 16 + lo];
    float* Crow = C + (size_t)(tm * 16 + hi * 8) * N + tn * 16 + lo;
#pragma unroll
    for (int i = 0; i < 8; ++i)                  // VGPR i -> M = i + 8*hi
        Crow[(size_t)i * N] = acc[i] + badd;
}

// ---------------- per-node attention logits ----------------
__global__ void gat_node_logits(const float* __restrict__ hfeat,
                                const float* __restrict__ a_src,
                                const float* __restrict__ a_dst,
                                float* __restrict__ as_out,
                                float* __restrict__ ad_out) {
    int idx = blockIdx.x * blockDim.x + threadIdx.x;
    if (idx >= N_NODES * HEADS) return;
    const int n = idx >> 2;
    const int hh = idx & 3;
    const float* hp = hfeat + (size_t)n * HC + hh * HIDDEN;
    const float* asv = a_src + hh * HIDDEN;
    const float* adv = a_dst + hh * HIDDEN;
    float s = 0.0f, d = 0.0f;
#pragma unroll 8
    for (int c = 0; c < HIDDEN; ++c) {
        float v = hp[c];
        s = fmaf(v, asv[c], s);
        d = fmaf(v, adv[c], d);
    }
    as_out[idx] = s;
    ad_out[idx] = d;
}

// ---------------- init: agg=0, m=-inf, denom=0 ----------------
__global__ void gat_init(float* __restrict__ agg, float* __restrict__ m,
                         float* __restrict__ denom) {
    int idx = blockIdx.x * blockDim.x + threadIdx.x;
    if (idx < N_NODES * HC) agg[idx] = 0.0f;
    if (idx < N_NODES * HEADS) {
        m[idx] = -INFINITY;
        denom[idx] = 0.0f;
    }
}

// sign-split float atomic max (int max for >=0, uint min for <0)
__device__ __forceinline__ void atomicMaxF32(float* addr, float val) {
    if (val >= 0.0f) {
        atomicMax((int*)addr, __float_as_int(val));
    } else {
        atomicMin((unsigned int*)addr, __float_as_uint(val));
    }
}

// ---------------- edge pass 1: leaky-relu score + segment max ----------
__global__ void gat_edge_score(const int* __restrict__ eidx,
                               const float* __restrict__ as_in,
                               const float* __restrict__ ad_in,
                               float* __restrict__ ebuf,
                               float* __restrict__ m) {
    int idx = blockIdx.x * blockDim.x + threadIdx.x;
    if (idx >= N_EDGES * HEADS) return;
    const int e  = idx >> 2;
    const int hh = idx & 3;
    const int src = eidx[e];
    const int dst = eidx[N_EDGES + e];
    float v = as_in[src * HEADS + hh] + ad_in[dst * HEADS + hh];
    v = (v >= 0.0f) ? v : NEG_SLOPE * v;   // LeakyReLU(0.2)
    ebuf[idx] = v;
    atomicMaxF32(&m[dst * HEADS + hh], v);
}

// ---------------- edge pass 2: exp(e - m[dst]) + segment sum ----------
__global__ void gat_edge_exp(const int* __restrict__ eidx,
                             const float* __restrict__ m,
                             float* __restrict__ ebuf,
                             float* __restrict__ denom) {
    int idx = blockIdx.x * blockDim.x + threadIdx.x;
    if (idx >= N_EDGES * HEADS) return;
    const int e  = idx >> 2;
    const int hh = idx & 3;
    const int dst = eidx[N_EDGES + e];
    float ex = expf(ebuf[idx] - m[dst * HEADS + hh]);
    ebuf[idx] = ex;
    atomicAdd(&denom[dst * HEADS + hh], ex);
}

// ---------------- edge pass 3: weighted scatter-add aggregation -------
// one thread per (edge, 4-channel quad): b128 gather + 4 f32 atomics
__global__ void gat_edge_aggregate(const int* __restrict__ eidx,
                                   const float* __restrict__ hfeat,
                                   const float* __restrict__ ebuf,
                                   const float* __restrict__ denom,
                                   float* __restrict__ agg) {
    long long idx = (long long)blockIdx.x * blockDim.x + threadIdx.x;
    if (idx >= (long long)N_EDGES * (HC / 4)) return;
    const int e = (int)(idx >> 5);           // / 32 quads per edge
    const int q = (int)(idx & 31);
    const int c = q * 4;
    const int hh = q >> 3;                   // head = c/32
    const int src = eidx[e];
    const int dst = eidx[N_EDGES + e];
    const float alpha = ebuf[e * HEADS + hh] /
                        (denom[dst * HEADS + hh] + 1e-16f);
    const float4 hv = *(const float4*)(hfeat + (size_t)src * HC + c);
    float* base = agg + (size_t)dst * HC + c;
    atomicAdd(base + 0, alpha * hv.x);
    atomicAdd(base + 1, alpha * hv.y);
    atomicAdd(base + 2, alpha * hv.z);
    atomicAdd(base + 3, alpha * hv.w);
}

// ---------------- epilogue: agg = elu(agg + bias) ----------------
__global__ void gat_bias_elu(float* __restrict__ agg,
                             const float* __restrict__ bias) {
    int idx = blockIdx.x * blockDim.x + threadIdx.x;
    if (idx >= N_NODES * HC) return;
    float v = agg[idx] + bias[idx & (HC - 1)];
    agg[idx] = (v > 0.0f) ? v : (expf(v) - 1.0f);
}

// ------------------------------------------------------------------
extern "C" void kernel_launch(void* const* d_in, const int* in_sizes, int n_in,
                              void* d_out, int out_size, void* d_ws, size_t ws_size,
                              hipStream_t stream) {
    const float* x     = (const float*)d_in[0];   // [N, 128]
    const int*   eidx  = (const int*)d_in[1];     // [2, E]
    // d_in[2] = edge_weight (unused by reference)
    const float* W     = (const float*)d_in[3];   // [128, 128]
    const float* a_src = (const float*)d_in[4];   // [4, 32]
    const float* a_dst = (const float*)d_in[5];   // [4, 32]
    const float* bias  = (const float*)d_in[6];   // [128]
    const float* W2    = (const float*)d_in[7];   // [128, 64]
    const float* b2    = (const float*)d_in[8];   // [64]
    float* out = (float*)d_out;                   // [N, 64]

    // workspace layout (floats)
    float* ws   = (float*)d_ws;
    float* hbuf = ws;                                   // N*128
    float* agg  = hbuf + (size_t)N_NODES * HC;          // N*128
    float* ebuf = agg  + (size_t)N_NODES * HC;          // E*4
    float* asb  = ebuf + (size_t)N_EDGES * HEADS;       // N*4
    float* adb  = asb  + (size_t)N_NODES * HEADS;       // N*4
    float* mb   = adb  + (size_t)N_NODES * HEADS;       // N*4
    float* den  = mb   + (size_t)N_NODES * HEADS;       // N*4

    const int TB = 256;
    const int tilesM    = N_NODES / 16;                 // 3125
    const int tilesMpad = (tilesM + 7) & ~7;            // 3128

    // 1) h = x @ W  (WMMA f32, N=128, K=128)
    {
        int blocks = (tilesMpad / 8) * (HC / 16);       // 391 * 8
        gat_gemm_f32_wmma<HC, F_IN, false>
            <<<blocks, TB, 0, stream>>>(x, W, nullptr, hbuf, N_NODES);
    }
    // 2) per-node logits
    gat_node_logits<<<(N_NODES * HEADS + TB - 1) / TB, TB, 0, stream>>>(
        hbuf, a_src, a_dst, asb, adb);
    // init m / denom / agg
    gat_init<<<(N_NODES * HC + TB - 1) / TB, TB, 0, stream>>>(agg, mb, den);
    // 3) edge scores + segment max
    gat_edge_score<<<(N_EDGES * HEADS + TB - 1) / TB, TB, 0, stream>>>(
        eidx, asb, adb, ebuf, mb);
    // 4) exp + segment sum
    gat_edge_exp<<<(N_EDGES * HEADS + TB - 1) / TB, TB, 0, stream>>>(
        eidx, mb, ebuf, den);
    // 5) weighted scatter-add aggregation (float4 per thread)
    {
        long long total = (long long)N_EDGES * (HC / 4);   // 25.6M
        int blocks = (int)((total + TB - 1) / TB);
        gat_edge_aggregate<<<blocks, TB, 0, stream>>>(eidx, hbuf, ebuf, den, agg);
    }
    // 6) bias + ELU
    gat_bias_elu<<<(N_NODES * HC + TB - 1) / TB, TB, 0, stream>>>(agg, bias);
    // 7) out = act @ W2 + b2  (WMMA f32, N=64, K=128)
    {
        int blocks = (tilesMpad / 8) * (F_OUT / 16);    // 391 * 4
        gat_gemm_f32_wmma<F_OUT, HC, true>
            <<<blocks, TB, 0, stream>>>(agg, W2, b2, out, N_NODES);
    }
}